// DDSPUnisonDetuneLayer_32993938767989
// MI455X (gfx1250) — compile-verified
//
#include <hip/hip_runtime.h>
#include <hip/hip_bf16.h>
#include <math.h>

// ---------------- problem constants ----------------
#define SR_F     48000.0f
#define T_LEN    62400
#define L_LEN    250
#define BATCH    16
#define NVOICE   16
#define D_LAT    128
#define C_COND   32
#define DIN      160   // D_LAT + C_COND

typedef __attribute__((ext_vector_type(2))) float v2f;
typedef __attribute__((ext_vector_type(8))) float v8f;

__device__ __forceinline__ float sigmoidf_(float x) {
    return 1.0f / (1.0f + __expf(-x));
}
__device__ __forceinline__ float softplusf_(float x) {
    // numerically-stable softplus matching jax.nn.softplus
    return fmaxf(x, 0.0f) + log1pf(__expf(-fabsf(x)));
}

// ---------------------------------------------------------------------------
// Kernel 1: pooled-latent MLP -> per-batch depth/spread/mask/norm (tiny)
// ---------------------------------------------------------------------------
__global__ __launch_bounds__(256) void mlp_kernel(
    const float* __restrict__ z, const float* __restrict__ cond,
    const float* __restrict__ W1, const float* __restrict__ b1,
    const float* __restrict__ W2, const float* __restrict__ b2,
    const float* __restrict__ W3, const float* __restrict__ b3,
    const float* __restrict__ W4, const float* __restrict__ b4,
    float* __restrict__ pdepth, float* __restrict__ pspread,
    float* __restrict__ pmask,  float* __restrict__ pnorm)
{
    __shared__ float sx[BATCH * DIN];     // 2560
    __shared__ float sh1[BATCH * 256];    // 4096
    __shared__ float sh2[BATCH * 128];    // 2048
    __shared__ float sh3[BATCH * 64];     // 1024
    __shared__ float sp[BATCH * 4];
    const int tid = threadIdx.x;

    // z_global = mean over L
    for (int idx = tid; idx < BATCH * D_LAT; idx += 256) {
        int b = idx >> 7, d = idx & 127;
        const float* zp = z + (size_t)(b * L_LEN) * D_LAT + d;
        float s = 0.f;
        for (int l = 0; l < L_LEN; ++l) s += zp[l * D_LAT];
        sx[b * DIN + d] = s * (1.0f / (float)L_LEN);
    }
    for (int idx = tid; idx < BATCH * C_COND; idx += 256) {
        int b = idx >> 5, c = idx & 31;
        sx[b * DIN + D_LAT + c] = cond[b * C_COND + c];
    }
    __syncthreads();

    for (int idx = tid; idx < BATCH * 256; idx += 256) {
        int b = idx >> 8, n = idx & 255;
        float acc = b1[n];
        for (int k = 0; k < DIN; ++k) acc += sx[b * DIN + k] * W1[k * 256 + n];
        sh1[b * 256 + n] = fmaxf(acc, 0.f);
    }
    __syncthreads();
    for (int idx = tid; idx < BATCH * 128; idx += 256) {
        int b = idx >> 7, n = idx & 127;
        float acc = b2[n];
        for (int k = 0; k < 256; ++k) acc += sh1[b * 256 + k] * W2[k * 128 + n];
        sh2[b * 128 + n] = fmaxf(acc, 0.f);
    }
    __syncthreads();
    for (int idx = tid; idx < BATCH * 64; idx += 256) {
        int b = idx >> 6, n = idx & 63;
        float acc = b3[n];
        for (int k = 0; k < 128; ++k) acc += sh2[b * 128 + k] * W3[k * 64 + n];
        sh3[b * 64 + n] = fmaxf(acc, 0.f);
    }
    __syncthreads();
    for (int idx = tid; idx < BATCH * 4; idx += 256) {
        int b = idx >> 2, n = idx & 3;
        float acc = b4[n];
        for (int k = 0; k < 64; ++k) acc += sh3[b * 64 + k] * W4[k * 4 + n];
        sp[idx] = acc;
    }
    __syncthreads();
    if (tid < BATCH) {
        const int b = tid;
        float nv     = 1.0f + 14.0f * sigmoidf_(sp[b * 4 + 0]);
        float spread = sigmoidf_(sp[b * 4 + 2]);
        float depth  = sigmoidf_(sp[b * 4 + 3]) * 0.5f;
        float msum = 0.f;
        for (int v = 0; v < NVOICE; ++v) {
            float m = sigmoidf_((nv - (float)v) * 2.0f);
            pmask[b * NVOICE + v] = m;
            msum += m;
        }
        pnorm[b]   = sqrtf(msum + 1e-6f);
        pdepth[b]  = depth;
        pspread[b] = spread;
    }
}

// ---------------------------------------------------------------------------
// Kernels 2-4: implicit-GEMM 'SAME' conv1d via V_WMMA_F32_16X16X4_F32.
// One wave computes one 16(M=positions) x 16(N=out-channels) tile.
// f32 WMMA layouts (ISA 7.12.2): A/B: M(or N)=lane&15, K pair = (lane>>4)*2;
// C/D: N=lane&15, rows (lane>>4)*8 + vgpr.
// ---------------------------------------------------------------------------
template<int KW, int CIN, int COUT, bool RELU, bool CONCAT>
__global__ __launch_bounds__(128) void conv_wmma_kernel(
    const float* __restrict__ in, const float* __restrict__ cond,
    const float* __restrict__ W,  const float* __restrict__ bias,
    float* __restrict__ out)
{
    const int wave = blockIdx.x * (blockDim.x >> 5) + (threadIdx.x >> 5);
    const int NT = COUT / 16;
    const int nt  = wave % NT;
    const int mt  = (wave / NT) % 16;
    const int b   = wave / (NT * 16);

    const int lane = threadIdx.x & 31;
    const int m    = lane & 15;          // A row (output position in tile) / B,N column
    const int koff = (lane >> 4) * 2;    // which K pair this half-wave holds
    const int n    = nt * 16 + m;        // output channel for B/D columns

    const float bn = bias[n];
    v8f acc = {bn, bn, bn, bn, bn, bn, bn, bn};

    const int l = mt * 16 + m;           // output position this lane's A rows map to

    for (int kw = 0; kw < KW; ++kw) {
        const int pos   = l + kw - KW / 2;
        const int posc  = min(max(pos, 0), L_LEN - 1);
        const float pv  = (pos >= 0 && pos < L_LEN) ? 1.0f : 0.0f;
        const float* wrow = W + (size_t)kw * CIN * COUT;
        const float* irow;
        if (CONCAT) irow = in + (size_t)(b * L_LEN + posc) * D_LAT;
        else        irow = in + (size_t)(b * L_LEN + posc) * CIN;

        #pragma unroll 4
        for (int c = 0; c < CIN; c += 4) {
            const int ca = c + koff;
            v2f a, bm;
            if (CONCAT) {
                // channels [0,128) come from z, [128,160) from broadcast cond
                float a0 = (ca < D_LAT) ? irow[ca]     : cond[b * C_COND + (ca - D_LAT)];
                float a1 = (ca + 1 < D_LAT) ? irow[ca + 1] : cond[b * C_COND + (ca + 1 - D_LAT)];
                a.x = a0 * pv;
                a.y = a1 * pv;
            } else {
                a.x = irow[ca] * pv;
                a.y = irow[ca + 1] * pv;
            }
            bm.x = wrow[(size_t)ca * COUT + n];
            bm.y = wrow[(size_t)(ca + 1) * COUT + n];
            acc = __builtin_amdgcn_wmma_f32_16x16x4_f32(
                false, a, false, bm, (short)0, acc, false, false);
        }
    }

    // store D: lanes 0-15 -> rows 0..7, lanes 16-31 -> rows 8..15 (col = n)
    const int rbase = mt * 16 + (lane >> 4) * 8;
    #pragma unroll
    for (int r = 0; r < 8; ++r) {
        const int row = rbase + r;
        if (row < L_LEN) {
            float v = acc[r];
            if (RELU) v = fmaxf(v, 0.f);
            out[(size_t)(b * L_LEN + row) * COUT + n] = v;
        }
    }
}

// ---------------------------------------------------------------------------
// Kernel 5: fused per-sample synthesis (interp + softplus + voices + norm)
// ---------------------------------------------------------------------------
__global__ __launch_bounds__(256) void synth_kernel(
    const float* __restrict__ base, const float* __restrict__ g3,
    const float* __restrict__ pdepth, const float* __restrict__ pspread,
    const float* __restrict__ pmask,  const float* __restrict__ pnorm,
    float* __restrict__ out)
{
    const int b = blockIdx.y;
    __shared__ float s_mask[NVOICE];
    __shared__ float s_dep, s_spr, s_nrm;
    if (threadIdx.x < NVOICE) s_mask[threadIdx.x] = pmask[b * NVOICE + threadIdx.x];
    if (threadIdx.x == 16) s_dep = pdepth[b];
    if (threadIdx.x == 17) s_spr = pspread[b];
    if (threadIdx.x == 18) s_nrm = pnorm[b];
    __syncthreads();

    const int t = blockIdx.x * 256 + threadIdx.x;
    if (t >= T_LEN) return;

    const float scale = (float)L_LEN / (float)T_LEN;
    float src = ((float)t + 0.5f) * scale - 0.5f;
    src = fminf(fmaxf(src, 0.0f), (float)(L_LEN - 1));
    const int i0 = (int)floorf(src);
    const int i1 = min(i0 + 1, L_LEN - 1);
    const float frac = src - (float)i0;

    float vg0[NVOICE], vg1[NVOICE];
    {
        const float4* q0 = reinterpret_cast<const float4*>(g3 + (size_t)(b * L_LEN + i0) * NVOICE);
        const float4* q1 = reinterpret_cast<const float4*>(g3 + (size_t)(b * L_LEN + i1) * NVOICE);
        #pragma unroll
        for (int i = 0; i < 4; ++i) {
            float4 u = q0[i];
            vg0[4*i] = u.x; vg0[4*i+1] = u.y; vg0[4*i+2] = u.z; vg0[4*i+3] = u.w;
            float4 w = q1[i];
            vg1[4*i] = w.x; vg1[4*i+1] = w.y; vg1[4*i+2] = w.z; vg1[4*i+3] = w.w;
        }
    }

    const float tsec = (float)t / SR_F;
    const float dep = s_dep, spr = s_spr;
    const float* bb = base + (size_t)b * T_LEN;

    float unison = 0.f, gain_sum = 0.f;
    #pragma unroll
    for (int v = 0; v < NVOICE; ++v) {
        const float pos = ((float)v - 7.5f) * (1.0f / 16.0f);
        const int shift = (int)truncf(pos * 20.0f);       // compile-time per v
        int idx = t - shift;
        if (idx >= T_LEN) idx -= T_LEN;
        if (idx < 0)      idx += T_LEN;
        const float sh = bb[idx];
        const float lfo = __sinf(2.0f * 3.14159265358979323846f *
                                 (3.0f + 0.3f * (float)v) * tsec);
        const float mod = sh * (1.0f + 0.2f * dep * lfo);
        const float vgain = softplusf_(vg0[v] * (1.0f - frac) + vg1[v] * frac);
        const float pan = 1.0f - fabsf(pos) * spr * 0.5f;
        unison   += mod * vgain * pan;
        gain_sum += vgain * s_mask[v];
    }
    out[(size_t)b * T_LEN + t] = gain_sum / (s_nrm + 1e-6f) * unison;
}

// ---------------------------------------------------------------------------
extern "C" void kernel_launch(void* const* d_in, const int* in_sizes, int n_in,
                              void* d_out, int out_size, void* d_ws, size_t ws_size,
                              hipStream_t stream) {
    const float* base = (const float*)d_in[0];
    const float* z    = (const float*)d_in[1];
    const float* cond = (const float*)d_in[2];
    // d_in[3] fundamental_freq: unused by the reference
    const float* W1 = (const float*)d_in[4];  const float* b1 = (const float*)d_in[5];
    const float* W2 = (const float*)d_in[6];  const float* b2 = (const float*)d_in[7];
    const float* W3 = (const float*)d_in[8];  const float* b3 = (const float*)d_in[9];
    const float* W4 = (const float*)d_in[10]; const float* b4 = (const float*)d_in[11];
    const float* K1 = (const float*)d_in[12]; const float* cb1 = (const float*)d_in[13];
    const float* K2 = (const float*)d_in[14]; const float* cb2 = (const float*)d_in[15];
    const float* K3 = (const float*)d_in[16]; const float* cb3 = (const float*)d_in[17];

    float* ws = (float*)d_ws;
    float* g1 = ws;                                   // 16*250*128 = 512000
    float* g2 = g1 + (size_t)BATCH * L_LEN * 128;     // 16*250*64  = 256000
    float* g3 = g2 + (size_t)BATCH * L_LEN * 64;      // 16*250*16  =  64000
    float* pdepth  = g3 + (size_t)BATCH * L_LEN * NVOICE;
    float* pspread = pdepth + BATCH;
    float* pmask   = pspread + BATCH;
    float* pnorm   = pmask + BATCH * NVOICE;

    mlp_kernel<<<1, 256, 0, stream>>>(z, cond, W1, b1, W2, b2, W3, b3, W4, b4,
                                      pdepth, pspread, pmask, pnorm);

    // waves = B * Mtiles * Ntiles; 4 waves per 128-thread block
    conv_wmma_kernel<7, DIN, 128, true,  true ><<<(BATCH*16*8)/4, 128, 0, stream>>>(z,  cond,    K1, cb1, g1);
    conv_wmma_kernel<5, 128,  64, true,  false><<<(BATCH*16*4)/4, 128, 0, stream>>>(g1, nullptr, K2, cb2, g2);
    conv_wmma_kernel<3, 64,   16, false, false><<<(BATCH*16*1)/4, 128, 0, stream>>>(g2, nullptr, K3, cb3, g3);

    dim3 gs((T_LEN + 255) / 256, BATCH);
    synth_kernel<<<gs, 256, 0, stream>>>(base, g3, pdepth, pspread, pmask, pnorm,
                                         (float*)d_out);
}